// EsophagealCancer_34488587387023
// MI455X (gfx1250) — compile-verified
//
#include <hip/hip_runtime.h>

// ---------------------------------------------------------------------------
// Esophageal-cancer Markov Monte Carlo, CDNA5 (gfx1250, wave32).
// One thread = one replicate path.
//
// Each transition-cumsum row is compressed to 3 breakpoints + constant
// weights; weights are pre-scaled by 32 so the accumulated count IS the next
// row's LDS byte offset.  Row layout (8 dwords, two B128 gathers per step):
//   [t0, t1, t2, life(current state) | w0*32, w1*32, w2*32, 0]
// Life accumulation is deferred to the next step's gather (exact: every
// skipped path has pending life == 0).  age>100 is block-uniform and hoisted.
// Block QALY reduction uses V_WMMA_F32_16X16X4_F32 with an all-ones A matrix.
// ---------------------------------------------------------------------------

typedef float v2f __attribute__((ext_vector_type(2)));
typedef float v8f __attribute__((ext_vector_type(8)));

#define N_YEARS 46
#define BLOCK   256

__constant__ float c_DYING[46] = {
    0.007779f, 0.008415f, 0.009074f, 0.009727f, 0.010371f, 0.011034f,
    0.011738f, 0.012489f, 0.013335f, 0.014319f, 0.015482f, 0.016824f,
    0.01833f,  0.0199f,   0.021539f, 0.023396f, 0.025476f, 0.027794f,
    0.03035f,  0.033204f, 0.036345f, 0.039788f, 0.04372f,  0.048335f,
    0.05365f,  0.059565f, 0.065848f, 0.072956f, 0.080741f, 0.089357f,
    0.09965f,  0.110901f, 0.123146f, 0.136412f, 0.15071f,  0.166038f,
    0.182374f, 0.199676f, 0.21788f,  0.236903f, 0.256636f, 0.276954f,
    0.297713f, 0.318755f, 0.339914f, 1.0f};

__global__ __launch_bounds__(BLOCK)
void EsoSim_kernel(const float* __restrict__ X,
                   const float* __restrict__ month,   // (n, R, 552)
                   const float* __restrict__ natm,    // (n, R, 46)
                   const float* __restrict__ dis,     // (n, R)
                   float* __restrict__ partials,      // (n, nblk)
                   int R)
{
    const int  tid   = threadIdx.x;
    const int  row   = blockIdx.y;
    const int  rep   = blockIdx.x * BLOCK + tid;
    const bool valid = rep < R;
    const int  repc  = valid ? rep : (R - 1);   // clamp: no OOB, zeroed at end

    // ---- per-row (block-uniform) parameters -------------------------------
    const float* x = X + row * 5;
    const int   treatment = (int)x[0];
    const int   start_age = (int)x[1];
    const float risk      = x[2];
    const float aspirin   = x[3];
    const float statin    = x[4];

    const float p1  = 1.0f - powf(1.0f - risk, 1.0f / 12.0f);
    int ti = treatment - 1; ti = ti < 0 ? 0 : (ti > 1 ? 1 : ti);
    const float drug_factor = (treatment == 1) ? (1.0f - aspirin) : (1.0f - statin);
    const float ac  = (ti == 0) ? 0.0024f : 0.001f;
    const float kk  = risk * drug_factor / ac;
    const float p4t = (1.0f - powf(1.0f - risk * drug_factor - ac, 1.0f / 12.0f))
                      / (1.0f + kk);
    const bool  treated = treatment > 0;
    const float p4  = treated ? p4t : 0.0f;
    const float p11 = treated ? kk * p4t : 0.0f;
    const float p5  = treated ? ((ti == 0) ? 0.9576f : 0.998f) : 0.0f;
    const float P3  = 0.02813745f;   // 1 - 0.71^(1/12)
    const float SEN = 2.0f;          // sentinel threshold: never < u (u < 1)

    // ---- compressed transition table in LDS (8 rows x 8 dwords) -----------
    __shared__ __align__(16) unsigned s_tab[64];
    __shared__ float s_q[BLOCK];

    if (tid == 0) {
        // helper-free explicit fill: [t0,t1,t2,lv | w0*32,w1*32,w2*32,0]
        // row 0 (state 1, lv 1.0): count = 1*(1-p1<u)
        s_tab[0] = __float_as_uint(1.0f - p1);
        s_tab[1] = __float_as_uint(SEN);
        s_tab[2] = __float_as_uint(SEN);
        s_tab[3] = __float_as_uint(1.0f);
        s_tab[4] = 1 * 32; s_tab[5] = 0; s_tab[6] = 0; s_tab[7] = 0;
        // row 1 (state 2, lv 0.5): count = 2*(0<u) + 1*(p2<u)   (p2 per-year)
        s_tab[8]  = __float_as_uint(0.0f);
        s_tab[9]  = __float_as_uint(SEN);          // p2, rewritten yearly
        s_tab[10] = __float_as_uint(SEN);
        s_tab[11] = __float_as_uint(0.5f);
        s_tab[12] = 2 * 32; s_tab[13] = 1 * 32; s_tab[14] = 0; s_tab[15] = 0;
        // row 2 (state 3, lv 0.5): count = 3*(0<u)+(0.8<u)+(0.96<u)
        s_tab[16] = __float_as_uint(0.0f);
        s_tab[17] = __float_as_uint(0.8f);
        s_tab[18] = __float_as_uint(0.96f);
        s_tab[19] = __float_as_uint(0.5f);
        s_tab[20] = 3 * 32; s_tab[21] = 1 * 32; s_tab[22] = 1 * 32; s_tab[23] = 0;
        // row 3 (state 4, lv 0.97): count = 3*(0<u)
        s_tab[24] = __float_as_uint(0.0f);
        s_tab[25] = __float_as_uint(SEN);
        s_tab[26] = __float_as_uint(SEN);
        s_tab[27] = __float_as_uint(0.97f);
        s_tab[28] = 3 * 32; s_tab[29] = 0; s_tab[30] = 0; s_tab[31] = 0;
        // row 4 (state 5, lv 0.5): count = 4*(0<u) + (1-P3<u)
        s_tab[32] = __float_as_uint(0.0f);
        s_tab[33] = __float_as_uint(1.0f - P3);
        s_tab[34] = __float_as_uint(SEN);
        s_tab[35] = __float_as_uint(0.5f);
        s_tab[36] = 4 * 32; s_tab[37] = 1 * 32; s_tab[38] = 0; s_tab[39] = 0;
        // row 5 (state 6, lv 0.0): count = 5*(0<u)
        s_tab[40] = __float_as_uint(0.0f);
        s_tab[41] = __float_as_uint(SEN);
        s_tab[42] = __float_as_uint(SEN);
        s_tab[43] = __float_as_uint(0.0f);
        s_tab[44] = 5 * 32; s_tab[45] = 0; s_tab[46] = 0; s_tab[47] = 0;
        // row 6 (state 7, lv 1.0): count = (0<u) + 5*(p11<u) + (1-p4<u)
        s_tab[48] = __float_as_uint(0.0f);
        s_tab[49] = __float_as_uint(p11);
        s_tab[50] = __float_as_uint(1.0f - p4);
        s_tab[51] = __float_as_uint(1.0f);
        s_tab[52] = 1 * 32; s_tab[53] = 5 * 32; s_tab[54] = 1 * 32; s_tab[55] = 0;
        // row 7 (state 8, lv 1.0): count = 5*(p5<u)
        s_tab[56] = __float_as_uint(p5);
        s_tab[57] = __float_as_uint(SEN);
        s_tab[58] = __float_as_uint(SEN);
        s_tab[59] = __float_as_uint(1.0f);
        s_tab[60] = 5 * 32; s_tab[61] = 0; s_tab[62] = 0; s_tab[63] = 0;
    }

    // ---- per-thread streams -----------------------------------------------
    const float*  mu  = month + ((size_t)row * R + repc) * 552;  // 16B aligned
    const float4* mu4 = (const float4*)mu;
    const float*  nu  = natm + ((size_t)row * R + repc) * 46;
    const float   du  = dis[(size_t)row * R + repc];

    int   soff   = treated ? 6 * 32 : 0;   // (state-1)*32: LDS row byte offset
    float lsum   = 0.0f;                   // life of col 0 added by 1st gather
    float lsum8  = 0.0f;
    float seen8f = 0.0f;

    // years with age > 100 contribute exactly zero (block-uniform) -> hoist
    int ysim = 101 - start_age;
    if (ysim > N_YEARS) ysim = N_YEARS;

    for (int y = 0; y < ysim; ++y) {
        const int age = start_age + y;
        __syncthreads();                                  // table no longer in use
        if (tid == 0) s_tab[9] = __float_as_uint(-0.0023f * (float)age + 1.1035f);
        __syncthreads();

        // skip whole years once every lane in the wave is dead (state 6)
        if (__ballot(soff != 5 * 32) != 0ull) {
            const float4 ua = mu4[y * 3 + 0];
            const float4 ub = mu4[y * 3 + 1];
            const float4 uc = mu4[y * 3 + 2];
            if (y + 1 < N_YEARS) __builtin_prefetch(&mu4[(y + 1) * 3], 0, 0);

            const int   yi   = age - 55;                  // in [0,45] while age<=100
            const float dyv  = c_DYING[yi];
            const bool  deathp = (nu[yi] <= dyv);         // hoisted: year-invariant

            const float uv[12] = {ua.x, ua.y, ua.z, ua.w,
                                  ub.x, ub.y, ub.z, ub.w,
                                  uc.x, uc.y, uc.z, uc.w};
            #pragma unroll
            for (int m = 0; m < 12; ++m) {
                const float u = uv[m];
                const char* rowp = (const char*)s_tab + soff;
                const uint4 tl = *(const uint4*)rowp;        // t0,t1,t2,lv
                const uint4 wv = *(const uint4*)(rowp + 16); // w0*32,w1*32,w2*32,0
                const float lv = __uint_as_float(tl.w);      // life(current state)
                lsum  += lv;                                 // deferred column add
                lsum8  = fmaf(seen8f, lv, lsum8);
                int cnt = ((__uint_as_float(tl.x) < u) ? (int)wv.x : 0)
                        + ((__uint_as_float(tl.y) < u) ? (int)wv.y : 0)
                        + ((__uint_as_float(tl.z) < u) ? (int)wv.z : 0);
                if (m == 5)                                  // natural mortality
                    cnt = (deathp && cnt != 5 * 32) ? 5 * 32 : cnt;
                soff = cnt;                                  // next (state-1)*32
                seen8f = (soff == 7 * 32) ? 1.0f : seen8f;   // latch state 8
            }
        }
    }

    // final column's life value (one last gather of the lv slot)
    {
        const float lv = __uint_as_float(s_tab[(soff >> 2) + 3]);
        lsum  += lv;
        lsum8  = fmaf(seen8f, lv, lsum8);
    }

    // ---- per-replicate QALY ----------------------------------------------
    float q = lsum;
    const bool apply_dis = (seen8f != 0.0f) && (du <= 0.058f) && (treatment == 1);
    if (apply_dis) q -= 0.39f * lsum8;       // life*0.61 from first state-8 on
    q *= (1.0f / 12.0f);
    if (!valid) q = 0.0f;

    s_q[tid] = q;
    __syncthreads();

    // ---- block reduction via V_WMMA_F32_16X16X4_F32 -----------------------
    // A = ones(16x4)  =>  D[m][n] = sum_k B[k][n]; C accumulates 4 chunks of
    // 64 values each; then a 4-step butterfly over the 16 column sums.
    if (tid < 32) {     // wave 0, EXEC all ones as WMMA requires
        v2f a; a[0] = 1.0f; a[1] = 1.0f;
        v8f acc = {0.0f, 0.0f, 0.0f, 0.0f, 0.0f, 0.0f, 0.0f, 0.0f};
        #pragma unroll
        for (int c = 0; c < 4; ++c) {
            v2f b;
            b[0] = s_q[c * 64 + tid];
            b[1] = s_q[c * 64 + 32 + tid];
            acc = __builtin_amdgcn_wmma_f32_16x16x4_f32(
                /*neg_a=*/false, a, /*neg_b=*/false, b,
                /*c_mod=*/(short)0, acc, /*reuse_a=*/false, /*reuse_b=*/false);
        }
        float v = acc[0];                    // lane L: column sum for n = L%16
        v += __shfl_xor(v, 1);
        v += __shfl_xor(v, 2);
        v += __shfl_xor(v, 4);
        v += __shfl_xor(v, 8);               // lanes 0..15: total of 256 QALYs
        if (tid == 0) partials[blockIdx.y * gridDim.x + blockIdx.x] = v;
    }
}

__global__ void EsoFinalize_kernel(const float* __restrict__ partials,
                                   float* __restrict__ out,
                                   int nblk, int n, float invR)
{
    const int r = blockIdx.x * blockDim.x + threadIdx.x;
    if (r < n) {
        float s = 0.0f;
        for (int b = 0; b < nblk; ++b) s += partials[r * nblk + b];  // fixed order
        out[r] = s * invR;
    }
}

extern "C" void kernel_launch(void* const* d_in, const int* in_sizes, int n_in,
                              void* d_out, int out_size, void* d_ws, size_t ws_size,
                              hipStream_t stream)
{
    const float* X     = (const float*)d_in[0];   // (n,5)
    const float* month = (const float*)d_in[1];   // (n,R,552)
    const float* natm  = (const float*)d_in[2];   // (n,R,46)
    const float* dis   = (const float*)d_in[3];   // (n,R)
    float*       out   = (float*)d_out;           // (n,)

    const int n    = in_sizes[0] / 5;
    const int R    = in_sizes[3] / n;
    const int nblk = (R + BLOCK - 1) / BLOCK;

    float* partials = (float*)d_ws;               // n*nblk floats, fully rewritten

    dim3 grid(nblk, n);
    EsoSim_kernel<<<grid, BLOCK, 0, stream>>>(X, month, natm, dis, partials, R);
    EsoFinalize_kernel<<<1, 64, 0, stream>>>(partials, out, nblk, n,
                                             1.0f / (float)R);
}